// LinkPredictor_10428180595499
// MI455X (gfx1250) — compile-verified
//
#include <hip/hip_runtime.h>

#define NNODES 100000
#define NEDGES 1600000
#define INC  128
#define HIDC 128
#define OUTC 64

typedef __attribute__((ext_vector_type(2))) float v2f;
typedef __attribute__((ext_vector_type(8))) float v8f;

__device__ __forceinline__ void atomic_add_f32(float* addr, float val) {
    __hip_atomic_fetch_add(addr, val, __ATOMIC_RELAXED, __HIP_MEMORY_SCOPE_AGENT);
}

// CDNA5 async global->LDS copy: per-lane 16B chunk, tracked with ASYNCcnt.
__device__ __forceinline__ void async_copy_b128(unsigned int lds_byte_off, const void* gaddr) {
    asm volatile("global_load_async_to_lds_b128 %0, %1, off"
                 :: "v"(lds_byte_off), "v"(gaddr)
                 : "memory");
}

__device__ __forceinline__ void wait_asynccnt0() {
#if __has_builtin(__builtin_amdgcn_s_wait_asynccnt)
    __builtin_amdgcn_s_wait_asynccnt(0);
#else
    asm volatile("s_wait_asynccnt 0x0" ::: "memory");
#endif
}

// ---------------- init: deg = 1.0 (self loop), agg1 = 0, out = 0 ----------------
__global__ void gcn_init_kernel(float* __restrict__ deg, float* __restrict__ agg1,
                                float* __restrict__ outbuf) {
    size_t i = (size_t)blockIdx.x * blockDim.x + threadIdx.x;
    if (i < (size_t)NNODES * HIDC) agg1[i] = 0.0f;
    if (i < (size_t)NNODES * OUTC) outbuf[i] = 0.0f;
    if (i < (size_t)NNODES)        deg[i]   = 1.0f;
}

// ---------------- degree: deg[col[e]] += 1 ----------------
__global__ void gcn_degree_kernel(const int* __restrict__ col, float* __restrict__ deg) {
    int e = blockIdx.x * blockDim.x + threadIdx.x;
    if (e < NEDGES) atomic_add_f32(&deg[col[e]], 1.0f);
}

// ---------------- dinv = rsqrt(deg), in place (deg >= 1 always) ----------------
__global__ void gcn_dinv_kernel(float* __restrict__ deg) {
    int n = blockIdx.x * blockDim.x + threadIdx.x;
    if (n < NNODES) deg[n] = rsqrtf(deg[n]);
}

// ---------------- norm[e] = dinv[row[e]] * dinv[col[e]] ----------------
__global__ void gcn_norm_kernel(const int* __restrict__ row, const int* __restrict__ col,
                                const float* __restrict__ dinv, float* __restrict__ nrm) {
    int e = blockIdx.x * blockDim.x + threadIdx.x;
    if (e < NEDGES) nrm[e] = dinv[row[e]] * dinv[col[e]];
}

// ---------------- GEMM: out[N x C] = A[N x K] @ W[K x C], fp32 WMMA ----------------
// 256 threads = 8 waves; each wave owns a 16-row x C stripe; block covers 128 rows.
// Per 64-deep K phase:
//   - A tile (128 x 64, 32 KB) staged via async global->LDS b128 copies (coalesced)
//   - W chunk staged pair-swizzled: lwp[(k/2)*C + col] = {W[k][col], W[k+1][col]}
// so every WMMA operand is one aligned ds_load_b64 into an even VGPR pair.
template <int K, int C>
__global__ __launch_bounds__(256) void gcn_gemm_wmma(const float* __restrict__ A,
                                                     const float* __restrict__ W,
                                                     float* __restrict__ out, int nrows) {
    constexpr int KT   = 64;
    constexpr int NB   = C / 16;
    constexpr int ROWS = 128;
    __shared__ float lda[ROWS * KT];      // A tile  [lrow][kk], 32 KB
    __shared__ float lwp[KT * C];         // W pairs [(kk/2)*C + col] -> float2, 32/16 KB

    const int tid  = threadIdx.x;
    const int wave = tid >> 5;
    const int lane = tid & 31;
    const int lo   = lane & 15;
    const int hi   = lane >> 4;           // 0: K pair {k0,k0+1} / M=j   1: {k0+2,k0+3} / M=j+8
    const int blockrow = blockIdx.x * ROWS;
    const int rowbase  = blockrow + wave * 16;

    const unsigned int lda_base = (unsigned int)(size_t)(void*)&lda[0];

    v8f acc[NB];
#pragma unroll
    for (int nb = 0; nb < NB; ++nb) acc[nb] = (v8f){0.f, 0.f, 0.f, 0.f, 0.f, 0.f, 0.f, 0.f};

    for (int kb = 0; kb < K; kb += KT) {
        __syncthreads();   // previous phase's LDS reads complete before overwrite

        // --- async stage A[blockrow .. +128) x [kb .. kb+KT) : 2048 x 16B chunks ---
        constexpr int CHUNKS = ROWS * KT * 4 / 16;   // 2048
        constexpr int CPR    = KT * 4 / 16;          // 16 chunks per row
#pragma unroll
        for (int ci = tid; ci < CHUNKS; ci += 256) {
            int lrow = ci / CPR;
            int part = ci % CPR;
            int gr = blockrow + lrow;
            if (gr >= nrows) gr = nrows - 1;         // clamp: valid memory, stores predicated later
            const void* gaddr = (const void*)(A + (size_t)gr * K + kb + part * 4);
            async_copy_b128(lda_base + (unsigned int)(ci * 16), gaddr);
        }

        // --- stage W chunk pair-swizzled ---
        constexpr int PAIRS = (KT / 2) * C;
#pragma unroll
        for (int i = tid; i < PAIRS; i += 256) {
            int p    = i / C;
            int colx = i % C;
            float2 w2;
            w2.x = W[(size_t)(kb + 2 * p) * C + colx];
            w2.y = W[(size_t)(kb + 2 * p + 1) * C + colx];
            *(float2*)&lwp[i * 2] = w2;
        }

        wait_asynccnt0();
        __syncthreads();

        // --- compute: 16 k-steps x NB WMMAs, all operands via single ds_load_b64 ---
        const float* arow = &lda[(wave * 16 + lo) * KT + 2 * hi];
#pragma unroll
        for (int k0 = 0; k0 < KT; k0 += 4) {
            const float2 af = *(const float2*)(arow + k0);
            v2f a; a.x = af.x; a.y = af.y;
#pragma unroll
            for (int nb = 0; nb < NB; ++nb) {
                const float2 bf = *(const float2*)&lwp[((k0 / 2 + hi) * C + nb * 16 + lo) * 2];
                v2f b; b.x = bf.x; b.y = bf.y;
                acc[nb] = __builtin_amdgcn_wmma_f32_16x16x4_f32(
                    false, a, false, b, (short)0, acc[nb], false, false);
            }
        }
    }

    // C/D layout: VGPR j -> rows {rowbase+j (lanes 0-15), rowbase+j+8 (lanes 16-31)}
#pragma unroll
    for (int nb = 0; nb < NB; ++nb) {
#pragma unroll
        for (int j = 0; j < 8; ++j) {
            int rr = rowbase + j + 8 * hi;
            if (rr < nrows) out[(size_t)rr * C + nb * 16 + lo] = acc[nb][j];
        }
    }
}

// ---------------- scatter: agg[col[e]][c] += norm[e] * t[row[e]][c] ----------------
template <int C, int EPB>
__global__ void gcn_scatter_kernel(const int* __restrict__ row, const int* __restrict__ col,
                                   const float* __restrict__ nrm, const float* __restrict__ t,
                                   float* __restrict__ agg) {
    const int c = threadIdx.x;                       // 0..C-1, coalesced channel
    const long ebase = (long)blockIdx.x * EPB;
#pragma unroll
    for (int i = 0; i < EPB; ++i) {
        long e = ebase + i;
        if (e < NEDGES) {
            int u = row[e], v = col[e];
            float w = nrm[e];
            atomic_add_f32(&agg[(size_t)v * C + c], w * t[(size_t)u * C + c]);
        }
    }
}

// ---------------- epilogue 1: h = relu(agg1 + dinv^2 * t1 + b1), in place ----------------
__global__ void gcn_epi1_kernel(const float* __restrict__ t1, const float* __restrict__ dinv,
                                const float* __restrict__ b1, float* __restrict__ agg1) {
    size_t i = (size_t)blockIdx.x * blockDim.x + threadIdx.x;
    if (i < (size_t)NNODES * HIDC) {
        int v = (int)(i / HIDC);
        int c = (int)(i % HIDC);
        float d = dinv[v];
        float val = agg1[i] + d * d * t1[i] + b1[c];
        agg1[i] = val > 0.0f ? val : 0.0f;
    }
}

// ---------------- epilogue 2: out += dinv^2 * t2 + b2 ----------------
__global__ void gcn_epi2_kernel(const float* __restrict__ t2, const float* __restrict__ dinv,
                                const float* __restrict__ b2, float* __restrict__ outbuf) {
    size_t i = (size_t)blockIdx.x * blockDim.x + threadIdx.x;
    if (i < (size_t)NNODES * OUTC) {
        int v = (int)(i / OUTC);
        int c = (int)(i % OUTC);
        float d = dinv[v];
        outbuf[i] = outbuf[i] + d * d * t2[i] + b2[c];
    }
}

extern "C" void kernel_launch(void* const* d_in, const int* in_sizes, int n_in,
                              void* d_out, int out_size, void* d_ws, size_t ws_size,
                              hipStream_t stream) {
    const float* x   = (const float*)d_in[0];
    const int*   ei  = (const int*)d_in[1];
    const int*   row = ei;             // edge_index[0], source
    const int*   col = ei + NEDGES;    // edge_index[1], target
    const float* W1  = (const float*)d_in[2];
    const float* b1  = (const float*)d_in[3];
    const float* W2  = (const float*)d_in[4];
    const float* b2  = (const float*)d_in[5];
    float* out = (float*)d_out;

    // workspace layout (floats): deg/dinv | norm | t1 | agg1(h) | t2
    float* ws   = (float*)d_ws;
    float* deg  = ws;                                   // N          (becomes dinv in place)
    float* nrm  = deg + NNODES;                         // E
    float* t1   = nrm + NEDGES;                         // N*128
    float* agg1 = t1 + (size_t)NNODES * HIDC;           // N*128
    float* t2   = agg1 + (size_t)NNODES * HIDC;         // N*64

    {
        size_t total = (size_t)NNODES * HIDC;
        int blocks = (int)((total + 255) / 256);
        gcn_init_kernel<<<blocks, 256, 0, stream>>>(deg, agg1, out);
    }
    gcn_degree_kernel<<<(NEDGES + 255) / 256, 256, 0, stream>>>(col, deg);
    gcn_dinv_kernel<<<(NNODES + 255) / 256, 256, 0, stream>>>(deg);
    gcn_norm_kernel<<<(NEDGES + 255) / 256, 256, 0, stream>>>(row, col, deg, nrm);

    // layer 1
    gcn_gemm_wmma<INC, HIDC><<<(NNODES + 127) / 128, 256, 0, stream>>>(x, W1, t1, NNODES);
    gcn_scatter_kernel<HIDC, 8><<<(NEDGES + 7) / 8, HIDC, 0, stream>>>(row, col, nrm, t1, agg1);
    {
        size_t total = (size_t)NNODES * HIDC;
        gcn_epi1_kernel<<<(int)((total + 255) / 256), 256, 0, stream>>>(t1, deg, b1, agg1);
    }

    // layer 2
    gcn_gemm_wmma<HIDC, OUTC><<<(NNODES + 127) / 128, 256, 0, stream>>>(agg1, W2, t2, NNODES);
    gcn_scatter_kernel<OUTC, 8><<<(NEDGES + 7) / 8, OUTC, 0, stream>>>(row, col, nrm, t2, out);
    {
        size_t total = (size_t)NNODES * OUTC;
        gcn_epi2_kernel<<<(int)((total + 255) / 256), 256, 0, stream>>>(t2, deg, b2, out);
    }
}